// GATFusionNet_44659069944073
// MI455X (gfx1250) — compile-verified
//
#include <hip/hip_runtime.h>
#include <math.h>

// ---------------------------------------------------------------------------
// GATFusionNet for MI455X (gfx1250, wave32).  fp32 throughout; dense layers
// use V_WMMA_F32_16X16X4_F32 with the B panel staged in LDS via
// GLOBAL_LOAD_ASYNC_TO_LDS_B128 (+ s_wait_asynccnt).  Edge softmax and
// aggregation via global atomics.
// ---------------------------------------------------------------------------

typedef __attribute__((ext_vector_type(2))) float v2f;
typedef __attribute__((ext_vector_type(8))) float v8f;

static constexpr int Nn  = 20000;   // nodes
static constexpr int Ee  = 320000;  // edges (before self loops)
static constexpr int Bb  = 8;
static constexpr int Tt  = 12;
static constexpr int Ff  = 16;
static constexpr int GFD = 128;
static constexpr int HID = 64;
static constexpr int Hh  = 4;
static constexpr int HXC = Hh * HID;         // 256
static constexpr int ET  = Ee + Nn;          // edges incl. self loops
static constexpr int KMAX = 128;             // largest GEMM K

// ---------------------------------------------------------------------------
// Generic fp32 WMMA GEMM:  C[M,Nc] = act(A[M,K] @ B[K,Nc] + bias)
// One wave per 16x16 C tile; 4 waves per block (4 M-tiles, same N tile).
// B panel [K,16] for this N tile is async-copied to LDS once per block.
// A-frag (16x4 f32, ISA 7.12.2): lane&15 = M row, lane>>4 selects K pair.
// B-frag mirrored (lane&15 = N col).  C/D: VGPR v -> M = v + 8*(lane>>4).
// ---------------------------------------------------------------------------
__global__ void wmma_gemm_f32(const float* __restrict__ A, int lda,
                              const float* __restrict__ B, int ldb,
                              float* __restrict__ C, int ldc,
                              int M, int Ncols, int K,
                              const float* __restrict__ bias, int do_relu) {
    __shared__ float Bs[KMAX * 16];              // 8 KB max (K <= 128)

    const int ntile = blockIdx.x;

    // ---- cooperative async copy of the B panel: K rows x 16 cols, as
    //      16-byte chunks (4 chunks per row).  LDS layout: Bs[kk*16 + nn].
    {
        const unsigned bs_base = (unsigned)(uintptr_t)(void*)&Bs[0]; // low 32b = LDS offset
        const int nchunks = K * 4;               // K*16 floats / 4 per chunk
        for (int idx = threadIdx.x; idx < nchunks; idx += blockDim.x) {
            const int kk = idx >> 2;
            const int n4 = (idx & 3) << 2;
            const float* gsrc = B + (size_t)kk * ldb + ntile * 16 + n4;
            const unsigned lds_addr = bs_base + (unsigned)idx * 16u;
            asm volatile("global_load_async_to_lds_b128 %0, %1, off"
                         :: "v"(lds_addr), "v"(gsrc)
                         : "memory");
        }
        asm volatile("s_wait_asynccnt 0x0" ::: "memory");
        __syncthreads();
    }

    const int lane  = threadIdx.x & 31;
    const int wave  = threadIdx.x >> 5;
    const int mtile = blockIdx.y * 4 + wave;
    if (mtile * 16 >= M) return;                 // wave-uniform exit (EXEC stays all-1)

    const int r  = lane & 15;                    // M row (A) / N col (B) within tile
    const int kb = (lane >> 4) << 1;             // 0 or 2: which K pair this half-wave owns

    const float* __restrict__ Arow = A + (size_t)(mtile * 16 + r) * lda;

    v8f acc = {0.f, 0.f, 0.f, 0.f, 0.f, 0.f, 0.f, 0.f};
    for (int k0 = 0; k0 < K; k0 += 4) {
        __builtin_prefetch(Arow + k0 + 32, 0, 3);   // global_prefetch_b8 (near)
        v2f a, b;
        a.x = Arow[k0 + kb];
        a.y = Arow[k0 + kb + 1];
        b.x = Bs[(k0 + kb) * 16 + r];
        b.y = Bs[(k0 + kb + 1) * 16 + r];
        acc = __builtin_amdgcn_wmma_f32_16x16x4_f32(
            /*neg_a=*/false, a, /*neg_b=*/false, b,
            /*c_mod=*/(short)0, acc, /*reuse_a=*/false, /*reuse_b=*/false);
    }

    const int mbase = mtile * 16 + 8 * (lane >> 4);
    const int ncol  = ntile * 16 + (lane & 15);
    const float bv  = bias ? bias[ncol] : 0.f;
#pragma unroll
    for (int v = 0; v < 8; ++v) {
        float val = acc[v] + bv;
        if (do_relu) val = fmaxf(val, 0.f);
        C[(size_t)(mbase + v) * ldc + ncol] = val;
    }
}

// ---------------------------------------------------------------------------
// Small helper kernels
// ---------------------------------------------------------------------------
__global__ void fill_f32(float* __restrict__ p, float v, size_t n) {
    size_t i = (size_t)blockIdx.x * blockDim.x + threadIdx.x;
    size_t stride = (size_t)gridDim.x * blockDim.x;
    for (; i < n; i += stride) p[i] = v;
}

__global__ void reduce_sum_f32(const float* __restrict__ x, size_t n,
                               float* __restrict__ out) {
    size_t i = (size_t)blockIdx.x * blockDim.x + threadIdx.x;
    size_t stride = (size_t)gridDim.x * blockDim.x;
    float acc = 0.f;
    for (; i < n; i += stride) acc += x[i];
    atomicAdd(out, acc);
}

// wedot[h] = sum_c We[h*HID+c] * ae[h*HID+c]   (We is [1, H*HID])
__global__ void compute_wedot(const float* __restrict__ We,
                              const float* __restrict__ ae,
                              float* __restrict__ wedot) {
    int h = threadIdx.x;
    if (h >= Hh) return;
    float s = 0.f;
    for (int c = 0; c < HID; ++c) s += We[h * HID + c] * ae[h * HID + c];
    wedot[h] = s;
}

// s[n,h], d[n,h] attention projections of h-matrix [N, H*HID]
__global__ void compute_sd(const float* __restrict__ hmat,
                           const float* __restrict__ a_src,
                           const float* __restrict__ a_dst,
                           float* __restrict__ s, float* __restrict__ d) {
    int idx = blockIdx.x * blockDim.x + threadIdx.x;   // n*H + h
    if (idx >= Nn * Hh) return;
    int n = idx / Hh, h = idx % Hh;
    const float* hp = hmat + (size_t)n * HXC + h * HID;
    float ss = 0.f, dd = 0.f;
    for (int c = 0; c < HID; ++c) {
        float v = hp[c];
        ss += v * a_src[h * HID + c];
        dd += v * a_dst[h * HID + c];
    }
    s[idx] = ss;
    d[idx] = dd;
}

__device__ __forceinline__ void atomicMaxFloat(float* addr, float val) {
    if (val >= 0.f)
        atomicMax((int*)addr, __float_as_int(val));
    else
        atomicMin((unsigned int*)addr, __float_as_uint(val));
}

// pass 1: alpha = leaky_relu(s[src]+d[dst]+w*wedot); segment-max into amax[dst]
__global__ void edge_alpha_max(const int* __restrict__ src,
                               const int* __restrict__ dst,
                               const float* __restrict__ ew,
                               const float* __restrict__ msum,
                               const float* __restrict__ wedot,
                               const float* __restrict__ s,
                               const float* __restrict__ d,
                               float* __restrict__ alpha,
                               float* __restrict__ amax) {
    int e = blockIdx.x * blockDim.x + threadIdx.x;
    if (e >= ET) return;
    int si, di; float w;
    if (e < Ee) { si = src[e]; di = dst[e]; w = ew[e]; }
    else        { si = di = e - Ee; w = msum[0] * (1.0f / (float)Ee); }
#pragma unroll
    for (int h = 0; h < Hh; ++h) {
        float al = s[si * Hh + h] + d[di * Hh + h] + w * wedot[h];
        al = al > 0.f ? al : 0.2f * al;
        alpha[(size_t)e * Hh + h] = al;
        atomicMaxFloat(&amax[di * Hh + h], al);
    }
}

// pass 2: ex = exp(alpha - amax[dst]); denom[dst] += ex   (in-place alpha->ex)
__global__ void edge_exp_sum(const int* __restrict__ dst,
                             float* __restrict__ alpha,
                             const float* __restrict__ amax,
                             float* __restrict__ denom) {
    int e = blockIdx.x * blockDim.x + threadIdx.x;
    if (e >= ET) return;
    int di = (e < Ee) ? dst[e] : e - Ee;
#pragma unroll
    for (int h = 0; h < Hh; ++h) {
        float ex = __expf(alpha[(size_t)e * Hh + h] - amax[di * Hh + h]);
        alpha[(size_t)e * Hh + h] = ex;
        atomicAdd(&denom[di * Hh + h], ex);
    }
}

// pass 3: out[dst, h, c] += h[src, h, c] * ex/denom[dst,h]; one block per edge
__global__ void edge_aggregate(const int* __restrict__ src,
                               const int* __restrict__ dst,
                               const float* __restrict__ ex,
                               const float* __restrict__ denom,
                               const float* __restrict__ hmat,
                               float* __restrict__ out) {
    int e = blockIdx.x;
    int j = threadIdx.x;                       // 0..255 = h*64+c
    int si, di;
    if (e < Ee) { si = src[e]; di = dst[e]; } else { si = di = e - Ee; }
    int h = j >> 6;
    float coef = ex[(size_t)e * Hh + h] / denom[di * Hh + h];
    atomicAdd(&out[(size_t)di * HXC + j], hmat[(size_t)si * HXC + j] * coef);
}

// sp[n,c] = relu(mean_h agg[n,h,c] + bias[c])
__global__ void head_mean_bias_relu(const float* __restrict__ agg,
                                    const float* __restrict__ bias,
                                    float* __restrict__ sp) {
    int idx = blockIdx.x * blockDim.x + threadIdx.x;   // n*HID + c
    if (idx >= Nn * HID) return;
    int n = idx / HID, c = idx % HID;
    float v = 0.f;
#pragma unroll
    for (int h = 0; h < Hh; ++h) v += agg[(size_t)n * HXC + h * HID + c];
    sp[idx] = fmaxf(v * (1.0f / (float)Hh) + bias[c], 0.f);
}

// pack x_seq[:, T-1] -> xt[B*N, F] contiguous
__global__ void pack_xlast(const float* __restrict__ x, float* __restrict__ xt) {
    size_t total = (size_t)Bb * Nn * Ff;
    size_t i = (size_t)blockIdx.x * blockDim.x + threadIdx.x;
    if (i >= total) return;
    size_t b = i / ((size_t)Nn * Ff);
    size_t rem = i % ((size_t)Nn * Ff);
    xt[i] = x[b * (size_t)Tt * Nn * Ff + (size_t)(Tt - 1) * Nn * Ff + rem];
}

// fused = relu(tmp1 + broadcast(tmp2) + bf)   (in-place over tmp1)
__global__ void fuse_relu(float* __restrict__ fused,
                          const float* __restrict__ tmp2,
                          const float* __restrict__ bf) {
    size_t total = (size_t)Bb * Nn * HID;
    size_t i = (size_t)blockIdx.x * blockDim.x + threadIdx.x;
    if (i >= total) return;
    int c = (int)(i % HID);
    size_t r = i / HID;
    int n = (int)(r % Nn);
    float v = fused[i] + tmp2[(size_t)n * HID + c] + bf[c];
    fused[i] = fmaxf(v, 0.f);
}

// out[r] = hmid[r,:32] . Wd2 + bd2
__global__ void final_out(const float* __restrict__ hmid,
                          const float* __restrict__ Wd2,
                          const float* __restrict__ bd2,
                          float* __restrict__ out) {
    int r = blockIdx.x * blockDim.x + threadIdx.x;
    if (r >= Bb * Nn) return;
    float acc = bd2[0];
#pragma unroll
    for (int c = 0; c < 32; ++c) acc += hmid[(size_t)r * 32 + c] * Wd2[c];
    out[r] = acc;
}

// ---------------------------------------------------------------------------
static void launch_gemm(const float* A, int lda, const float* B, int ldb,
                        float* C, int ldc, int M, int Nc, int K,
                        const float* bias, int act, hipStream_t s) {
    dim3 grid(Nc / 16, (M / 16 + 3) / 4);
    wmma_gemm_f32<<<grid, 128, 0, s>>>(A, lda, B, ldb, C, ldc, M, Nc, K, bias, act);
}

extern "C" void kernel_launch(void* const* d_in, const int* in_sizes, int n_in,
                              void* d_out, int out_size, void* d_ws, size_t ws_size,
                              hipStream_t stream) {
    const float* x_seq = (const float*)d_in[0];
    const int*   eidx  = (const int*)d_in[1];      // [2,E]
    const float* ew    = (const float*)d_in[2];    // [E,1]
    const float* emb   = (const float*)d_in[3];    // [N,GFD]
    const float* Wt    = (const float*)d_in[4];
    const float* bt    = (const float*)d_in[5];
    const float* W1    = (const float*)d_in[6];
    const float* as1   = (const float*)d_in[7];
    const float* ad1   = (const float*)d_in[8];
    const float* We1   = (const float*)d_in[9];
    const float* ae1   = (const float*)d_in[10];
    const float* b1g   = (const float*)d_in[11];
    const float* W2    = (const float*)d_in[12];
    const float* as2   = (const float*)d_in[13];
    const float* ad2   = (const float*)d_in[14];
    const float* We2   = (const float*)d_in[15];
    const float* ae2   = (const float*)d_in[16];
    const float* b2g   = (const float*)d_in[17];
    const float* Wf    = (const float*)d_in[18];   // [128,64]
    const float* bf    = (const float*)d_in[19];
    const float* Wd1   = (const float*)d_in[20];   // [64,32]
    const float* bd1   = (const float*)d_in[21];
    const float* Wd2   = (const float*)d_in[22];   // [32,1]
    const float* bd2   = (const float*)d_in[23];

    const int* src = eidx;
    const int* dst = eidx + Ee;

    // ---- workspace carve-up (floats) ----
    float* W = (float*)d_ws;
    float* h1    = W;                    W += (size_t)Nn * HXC;      // 5.12M
    float* s_    = W;                    W += (size_t)Nn * Hh;
    float* dd_   = W;                    W += (size_t)Nn * Hh;
    float* alpha = W;                    W += (size_t)ET * Hh;
    float* amax  = W;                    W += (size_t)Nn * Hh;
    float* denom = W;                    W += (size_t)Nn * Hh;
    float* agg   = W;                    W += (size_t)Nn * HXC;      // 5.12M
    float* sp1   = W;                    W += (size_t)Nn * HID;
    float* sp2   = W;                    W += (size_t)Nn * HID;
    float* xt    = W;                    W += (size_t)Bb * Nn * Ff;
    float* xw    = W;                    W += (size_t)Bb * Nn * HID; // 10.24M
    float* tmp2  = W;                    W += (size_t)Nn * HID;
    float* fused = W;                    W += (size_t)Bb * Nn * HID; // 10.24M
    float* hmid  = W;                    W += (size_t)Bb * Nn * 32;
    float* wedot = W;                    W += 8;
    float* msum  = W;                    W += 8;

    const int TB = 256;
    auto blks = [](size_t n, int tb) { return (int)((n + tb - 1) / tb); };

    // mean(edge_weight): sum here, scaled inside edge_alpha_max
    hipMemsetAsync(msum, 0, sizeof(float), stream);
    reduce_sum_f32<<<256, TB, 0, stream>>>(ew, (size_t)Ee, msum);

    // ---------------- GAT layer 1 ----------------
    launch_gemm(emb, GFD, W1, HXC, h1, HXC, Nn, HXC, GFD, nullptr, 0, stream);
    compute_sd<<<blks((size_t)Nn * Hh, TB), TB, 0, stream>>>(h1, as1, ad1, s_, dd_);
    compute_wedot<<<1, 32, 0, stream>>>(We1, ae1, wedot);

    fill_f32<<<256, TB, 0, stream>>>(amax, -INFINITY, (size_t)Nn * Hh);
    hipMemsetAsync(denom, 0, (size_t)Nn * Hh * sizeof(float), stream);
    hipMemsetAsync(agg, 0, (size_t)Nn * HXC * sizeof(float), stream);

    edge_alpha_max<<<blks(ET, TB), TB, 0, stream>>>(src, dst, ew, msum, wedot, s_, dd_, alpha, amax);
    edge_exp_sum<<<blks(ET, TB), TB, 0, stream>>>(dst, alpha, amax, denom);
    edge_aggregate<<<ET, HXC, 0, stream>>>(src, dst, alpha, denom, h1, agg);
    head_mean_bias_relu<<<blks((size_t)Nn * HID, TB), TB, 0, stream>>>(agg, b1g, sp1);

    // ---------------- GAT layer 2 ----------------
    launch_gemm(sp1, HID, W2, HXC, h1, HXC, Nn, HXC, HID, nullptr, 0, stream);
    compute_sd<<<blks((size_t)Nn * Hh, TB), TB, 0, stream>>>(h1, as2, ad2, s_, dd_);
    compute_wedot<<<1, 32, 0, stream>>>(We2, ae2, wedot);

    fill_f32<<<256, TB, 0, stream>>>(amax, -INFINITY, (size_t)Nn * Hh);
    hipMemsetAsync(denom, 0, (size_t)Nn * Hh * sizeof(float), stream);
    hipMemsetAsync(agg, 0, (size_t)Nn * HXC * sizeof(float), stream);

    edge_alpha_max<<<blks(ET, TB), TB, 0, stream>>>(src, dst, ew, msum, wedot, s_, dd_, alpha, amax);
    edge_exp_sum<<<blks(ET, TB), TB, 0, stream>>>(dst, alpha, amax, denom);
    edge_aggregate<<<ET, HXC, 0, stream>>>(src, dst, alpha, denom, h1, agg);
    head_mean_bias_relu<<<blks((size_t)Nn * HID, TB), TB, 0, stream>>>(agg, b2g, sp2);

    // ---------------- temporal branch ----------------
    pack_xlast<<<blks((size_t)Bb * Nn * Ff, TB), TB, 0, stream>>>(x_seq, xt);
    launch_gemm(xt, Ff, Wt, HID, xw, HID, Bb * Nn, HID, Ff, bt, 0, stream);

    // ---------------- fusion + decoder ----------------
    // concat([xw, sp2]) @ Wf  ==  xw @ Wf[:64] + sp2 @ Wf[64:]
    launch_gemm(xw, HID, Wf, HID, fused, HID, Bb * Nn, HID, HID, nullptr, 0, stream);
    launch_gemm(sp2, HID, Wf + (size_t)HID * HID, HID, tmp2, HID, Nn, HID, HID, nullptr, 0, stream);
    fuse_relu<<<blks((size_t)Bb * Nn * HID, TB), TB, 0, stream>>>(fused, tmp2, bf);

    launch_gemm(fused, HID, Wd1, 32, hmid, 32, Bb * Nn, 32, HID, bd1, 1, stream);
    final_out<<<blks((size_t)Bb * Nn, TB), TB, 0, stream>>>(hmid, Wd2, bd2, (float*)d_out);
}